// GeneralDeformRoIPool_13469017440351
// MI455X (gfx1250) — compile-verified
//
#include <hip/hip_runtime.h>
#include <hip/hip_bf16.h>

namespace {
constexpr int   Bn    = 2;
constexpr int   Cn    = 256;
constexpr int   Hn    = 200;
constexpr int   Wn    = 200;
constexpr int   NROI  = 1000;
constexpr int   PHn   = 7;
constexpr int   PWn   = 7;
constexpr int   NBIN  = PHn * PWn;
constexpr float SCALE = 0.25f;
constexpr float GAMMA = 0.1f;
constexpr float SIGMA = 0.5f;
constexpr int   TX    = 32;   // x-tile width for transpose
constexpr int   LSTR  = 36;   // LDS row stride in words: 16B-aligned, gcd(36,64)=4 (2-way worst)
} // namespace

// ---- gfx1250 async-to-LDS path (guarded; falls back to ds_store_b128) ------
#if defined(__gfx1250__) && __has_builtin(__builtin_amdgcn_global_load_async_to_lds_b128)
#define USE_ASYNC_LDS 1
// Diagnostic from clang shows the builtin takes pointers to
// 'int __attribute__((vector_size(16)))' (GCC-style vector), global + LDS.
typedef int v4i_async __attribute__((vector_size(16)));
typedef __attribute__((address_space(1))) v4i_async* g_v4i_ptr;
typedef __attribute__((address_space(3))) v4i_async* l_v4i_ptr;
#else
#define USE_ASYNC_LDS 0
#endif

// =====================================================================
// Phase A: NCHW -> NHWC transpose through LDS tiles.
// Block = (x-tile, y, b); 256 threads. Reads coalesced along x,
// writes coalesced along c. Tile = 256 channels x 32 x-positions.
// =====================================================================
__global__ __launch_bounds__(256) void nchw_to_nhwc(const float* __restrict__ in,
                                                    float* __restrict__ nhwc) {
    __shared__ float tile[Cn * LSTR];

    const int t  = threadIdx.x;
    const int x0 = blockIdx.x * TX;
    const int y  = blockIdx.y;
    const int b  = blockIdx.z;

    const int xq   = (t & 7) * 4;   // 0,4,...,28 within the tile
    const int crow = t >> 3;        // 0..31 channel sub-row per pass
    const int xg   = x0 + xq;       // global x of this 16B group
    const bool grp_ok = (xg + 3) < Wn;   // W=200: 16B groups are all-or-nothing

#pragma unroll
    for (int p = 0; p < 8; ++p) {
        const int c = p * 32 + crow;
        const float* src  = in + (((size_t)b * Cn + c) * Hn + y) * Wn + xg;
        float*       ldst = &tile[c * LSTR + xq];
        if (grp_ok) {
#if USE_ASYNC_LDS
            __builtin_amdgcn_global_load_async_to_lds_b128(
                (g_v4i_ptr)(unsigned long long)(const void*)src,
                (l_v4i_ptr)(unsigned int)(unsigned long long)(void*)ldst,
                0, 0);
#else
            const float4 v = *(const float4*)src;
            *(float4*)ldst = v;   // ds_store_b128 (16B-aligned: LSTR % 4 == 0)
#endif
        }
    }

#if USE_ASYNC_LDS
#if __has_builtin(__builtin_amdgcn_s_wait_asynccnt)
    __builtin_amdgcn_s_wait_asynccnt(0);
#else
    asm volatile("s_wait_asynccnt 0x0" ::: "memory");
#endif
#endif
    __syncthreads();

    // gfx1250 global_prefetch_b8: warm next y-row of this channel into L2.
    if (y + 1 < Hn)
        __builtin_prefetch(in + (((size_t)b * Cn + t) * Hn + (y + 1)) * Wn + x0, 0, 1);

    // Write out: per iteration, 256 consecutive channels of one (y,x) -> 1KB coalesced.
    float* dst_row = nhwc + (((size_t)b * Hn + y) * Wn) * Cn;
#pragma unroll 4
    for (int xi = 0; xi < TX; ++xi) {
        const int xx = x0 + xi;
        if (xx < Wn)
            dst_row[(size_t)xx * Cn + t] = tile[t * LSTR + xi];
    }
}

// =====================================================================
// Phase B: one block per (roi, bin); 256 threads = channels.
// Threads 0..3 compute one sample each (mmcv DeformRoIPool rules) and
// publish 16 (row-offset, weight) pairs via LDS; weights fold in the
// 0.25 sample-average factor and the validity mask.
// =====================================================================
template <bool NHWC>
__global__ __launch_bounds__(256) void deform_pool(const float* __restrict__ feat,
                                                   const float* __restrict__ rois,
                                                   const float* __restrict__ resc,
                                                   const float* __restrict__ offs,
                                                   float* __restrict__ out) {
    __shared__ float s_w[16];
    __shared__ int   s_o[16];

    const int blk = blockIdx.x;
    const int n   = blk / NBIN;
    const int bin = blk % NBIN;
    const int ph  = bin / PWn;
    const int pw  = bin % PWn;
    const int t   = threadIdx.x;

    if (t < 4) {
        const float* r = rois + n * 5;
        const int   bI = (int)r[0];
        const float x1 = r[1] * SCALE - 0.5f;
        const float y1 = r[2] * SCALE - 0.5f;
        const float x2 = r[3] * SCALE - 0.5f;
        const float y2 = r[4] * SCALE - 0.5f;
        const float rw = x2 - x1, rh = y2 - y1;
        const float bw = rw / (float)PWn;
        const float bh = rh / (float)PHn;

        const int pidx = n * (2 * NBIN) + bin;        // plane 0 (x)
        const int pidy = pidx + NBIN;                 // plane 1 (y)
        const float offx = GAMMA * rw * offs[pidx];
        const float offy = GAMMA * rh * offs[pidy];
        const float sx   = 1.0f + SIGMA * resc[pidx]; // BETA = 0
        const float sy   = 1.0f + SIGMA * resc[pidy];

        const float cx = x1 + offx + ((float)pw + 0.5f) * bw;
        const float cy = y1 + offy + ((float)ph + 0.5f) * bh;

        const int a  = t >> 1;   // y sample index
        const int bb = t & 1;    // x sample index
        const float fra = ((float)a + 0.5f) * 0.5f;
        const float frb = ((float)bb + 0.5f) * 0.5f;
        const float xr = x1 + offx + ((float)pw + frb) * bw;
        const float yr = y1 + offy + ((float)ph + fra) * bh;
        const float X  = cx + (xr - cx) * sx;
        const float Y  = cy + (yr - cy) * sy;

        const bool valid = (Y > -1.0f) && (Y < (float)Hn) && (X > -1.0f) && (X < (float)Wn);
        const float yc = fminf(fmaxf(Y, 0.0f), (float)(Hn - 1));
        const float xc = fminf(fmaxf(X, 0.0f), (float)(Wn - 1));
        const float y0f = floorf(yc), x0f = floorf(xc);
        const int iy0 = (int)y0f, ix0 = (int)x0f;
        const int iy1 = min(iy0 + 1, Hn - 1);
        const int ix1 = min(ix0 + 1, Wn - 1);
        const float ly = yc - y0f, lx = xc - x0f;
        const float hy = 1.0f - ly, hx = 1.0f - lx;
        const float wsc = valid ? 0.25f : 0.0f;   // sample-average + validity

        const int base = bI * (Hn * Wn);
        const int o00 = NHWC ? (base + iy0 * Wn + ix0) * Cn : (base + iy0 * Wn + ix0);
        const int o01 = NHWC ? (base + iy0 * Wn + ix1) * Cn : (base + iy0 * Wn + ix1);
        const int o10 = NHWC ? (base + iy1 * Wn + ix0) * Cn : (base + iy1 * Wn + ix0);
        const int o11 = NHWC ? (base + iy1 * Wn + ix1) * Cn : (base + iy1 * Wn + ix1);

        s_w[t * 4 + 0] = hy * hx * wsc;  s_o[t * 4 + 0] = o00;
        s_w[t * 4 + 1] = hy * lx * wsc;  s_o[t * 4 + 1] = o01;
        s_w[t * 4 + 2] = ly * hx * wsc;  s_o[t * 4 + 2] = o10;
        s_w[t * 4 + 3] = ly * lx * wsc;  s_o[t * 4 + 3] = o11;

        if (NHWC) {  // gfx1250 prefetch: warm the 4 corner rows this thread owns
            __builtin_prefetch(feat + o00, 0, 1);
            __builtin_prefetch(feat + o01, 0, 1);
            __builtin_prefetch(feat + o10, 0, 1);
            __builtin_prefetch(feat + o11, 0, 1);
        }
    }
    __syncthreads();

    float acc = 0.0f;
    if (NHWC) {
#pragma unroll
        for (int k = 0; k < 16; ++k)
            acc += s_w[k] * feat[s_o[k] + t];          // 1KB contiguous row per k
    } else {
        const float* fc = feat + (size_t)t * (Hn * Wn);
#pragma unroll
        for (int k = 0; k < 16; ++k)
            acc += s_w[k] * fc[s_o[k]];
    }

    out[((size_t)n * Cn + t) * NBIN + bin] = acc;
}

// =====================================================================
// Launcher
// =====================================================================
extern "C" void kernel_launch(void* const* d_in, const int* in_sizes, int n_in,
                              void* d_out, int out_size, void* d_ws, size_t ws_size,
                              hipStream_t stream) {
    (void)in_sizes; (void)n_in; (void)out_size;
    const float* x    = (const float*)d_in[0];
    const float* rois = (const float*)d_in[1];
    const float* resc = (const float*)d_in[2];
    const float* offs = (const float*)d_in[3];
    float*       out  = (float*)d_out;

    const size_t nhwc_bytes = (size_t)Bn * Hn * Wn * Cn * sizeof(float);

    if (ws_size >= nhwc_bytes) {
        float* nhwc = (float*)d_ws;
        dim3 gA((Wn + TX - 1) / TX, Hn, Bn);
        nchw_to_nhwc<<<gA, 256, 0, stream>>>(x, nhwc);
        deform_pool<true><<<NROI * NBIN, 256, 0, stream>>>(nhwc, rois, resc, offs, out);
    } else {
        deform_pool<false><<<NROI * NBIN, 256, 0, stream>>>(x, rois, resc, offs, out);
    }
}